// ColorContrastLoss_44212393345413
// MI455X (gfx1250) — compile-verified
//
#include <hip/hip_runtime.h>

typedef __attribute__((ext_vector_type(2))) float v2f;
typedef __attribute__((ext_vector_type(8))) float v8f;
typedef __attribute__((ext_vector_type(4))) int   v4i;

#define TK      128      // K elements per LDS tile
#define MST     132      // LDS row stride (floats): %64 == 4 dwords -> conflict free
#define NWAVES  4
#define NMASK   32
#define TEMP_INV (1.0f / 0.07f)

#if __has_builtin(__builtin_amdgcn_global_load_async_to_lds_b128)
#define USE_ASYNC 1
#else
#define USE_ASYNC 0
#endif

typedef __attribute__((address_space(1))) v4i as1_v4i;   // global-space int4
typedef __attribute__((address_space(3))) v4i as3_v4i;   // LDS-space int4

__device__ __forceinline__ void wait_async_zero() {
#if USE_ASYNC
#if __has_builtin(__builtin_amdgcn_s_wait_asynccnt)
    __builtin_amdgcn_s_wait_asynccnt(0);
#else
    asm volatile("s_wait_asynccnt 0x0" ::: "memory");
#endif
#endif
}

// issue all global->LDS copies for one full tile (per wave: 8 mask rows + <=1 img row)
__device__ __forceinline__ void issue_tile(const float* __restrict__ mb,
                                           const float* __restrict__ ib,
                                           int HW, int k0,
                                           float* smask, float* simg,
                                           int w, int lane)
{
    const int col = lane * 4;
#if USE_ASYNC
    #pragma unroll
    for (int i = 0; i < 8; ++i) {
        const int row = i * 4 + w;
        __builtin_amdgcn_global_load_async_to_lds_b128(
            (as1_v4i*)(mb + (size_t)row * HW + k0 + col),
            (as3_v4i*)(smask + row * MST + col), 0, 0);
    }
    if (w < 3) {
        __builtin_amdgcn_global_load_async_to_lds_b128(
            (as1_v4i*)(ib + (size_t)w * HW + k0 + col),
            (as3_v4i*)(simg + w * MST + col), 0, 0);
    }
#else
    // fallback: stage in distinct registers first so 9 loads stay in flight
    float4 r[8];
    #pragma unroll
    for (int i = 0; i < 8; ++i) {
        const int row = i * 4 + w;
        r[i] = *(const float4*)(mb + (size_t)row * HW + k0 + col);
    }
    float4 ri;
    if (w < 3) ri = *(const float4*)(ib + (size_t)w * HW + k0 + col);
    #pragma unroll
    for (int i = 0; i < 8; ++i) {
        const int row = i * 4 + w;
        *(float4*)&smask[row * MST + col] = r[i];
    }
    if (w < 3) *(float4*)&simg[w * MST + col] = ri;
#endif
}

// 8 K-steps of V_WMMA_F32_16X16X4_F32 over this wave's K subrange of the tile
__device__ __forceinline__ void compute_tile(const float* smask, const float* sb,
                                             int w, int n, int half,
                                             v8f& acc0, v8f& acc1)
{
    #pragma unroll
    for (int j = 0; j < 8; ++j) {
        const int kk = w * 32 + j * 4 + 2 * half;
        const v2f bf = *(const v2f*)&sb[n * MST + kk];
        const v2f a0 = *(const v2f*)&smask[n * MST + kk];
        const v2f a1 = *(const v2f*)&smask[(n + 16) * MST + kk];
        acc0 = __builtin_amdgcn_wmma_f32_16x16x4_f32(false, a0, false, bf,
                                                     (short)0, acc0, false, false);
        acc1 = __builtin_amdgcn_wmma_f32_16x16x4_f32(false, a1, false, bf,
                                                     (short)0, acc1, false, false);
    }
}

// ---------------------------------------------------------------------------
// Main streaming kernel: colors[b][n][c] partial sums.
//   c = 0..2 : sum(mask * image_channel)   c = 3 : sum(mask)  (ones column)
// ---------------------------------------------------------------------------
__global__ void __launch_bounds__(128)
color_reduce_wmma(const float* __restrict__ masks,   // [B][32][HW]
                  const float* __restrict__ images,  // [B][3][HW]
                  float* __restrict__ partials,      // [B][nchunks][32][4]
                  int HW, int nTiles, int tilesPerChunk, int nchunks)
{
    __shared__ float sMask[2][NMASK * MST];
    __shared__ float sB[2][16 * MST];      // rows 0-2: RGB, row 3: ones, 4-15: zeros
    __shared__ float sRed[NWAVES * NMASK * 4];

    const int tid  = threadIdx.x;
    const int w    = tid >> 5;
    const int lane = tid & 31;
    const int b     = blockIdx.y;
    const int chunk = blockIdx.x;

    // one-time init of constant B rows in both buffers
    for (int idx = tid; idx < 13 * MST; idx += 128) {
        const float v = (idx < MST) ? 1.0f : 0.0f;
        sB[0][3 * MST + idx] = v;
        sB[1][3 * MST + idx] = v;
    }

    const float* mb = masks  + (size_t)b * NMASK * HW;
    const float* ib = images + (size_t)b * 3 * HW;

    v8f acc0 = {};   // mask rows 0..15
    v8f acc1 = {};   // mask rows 16..31

    const int tile0 = chunk * tilesPerChunk;
    int tile1 = tile0 + tilesPerChunk;
    if (tile1 > nTiles) tile1 = nTiles;

    const int nFullGlobal = HW / TK;                       // tiles with no bounds checks
    const int fullEnd = (tile1 < nFullGlobal) ? tile1 : nFullGlobal;

    const int n    = lane & 15;
    const int half = lane >> 4;

#if USE_ASYNC
    // software pipeline: copy tile t+1 while WMMAs chew on tile t
    if (tile0 < fullEnd)
        issue_tile(mb, ib, HW, tile0 * TK, sMask[0], sB[0], w, lane);
    for (int t = tile0; t < fullEnd; ++t) {
        const int cur = (t - tile0) & 1;
        wait_async_zero();          // this wave's copies into buf[cur] landed
        __syncthreads();            // everyone's copies landed; prev compute done
        if (t + 1 < fullEnd)
            issue_tile(mb, ib, HW, (t + 1) * TK, sMask[cur ^ 1], sB[cur ^ 1], w, lane);
        compute_tile(sMask[cur], sB[cur], w, n, half, acc0, acc1);
    }
#else
    for (int t = tile0; t < fullEnd; ++t) {
        // single LDS buffer: loads wait until every wave finished the previous tile
        __syncthreads();            // previous tile fully consumed
        issue_tile(mb, ib, HW, t * TK, sMask[0], sB[0], w, lane);
        __syncthreads();
        compute_tile(sMask[0], sB[0], w, n, half, acc0, acc1);
    }
#endif

    // at most one partial tail tile (only when HW % TK != 0)
    if (tile1 > nFullGlobal && tile0 <= nFullGlobal) {
        const int k0 = nFullGlobal * TK;
        __syncthreads();            // buffers free
        const int col = lane * 4;
        for (int i = 0; i < 8; ++i) {
            const int row = i * 4 + w;
            float4 v;
            v.x = (k0 + col + 0 < HW) ? mb[(size_t)row * HW + k0 + col + 0] : 0.0f;
            v.y = (k0 + col + 1 < HW) ? mb[(size_t)row * HW + k0 + col + 1] : 0.0f;
            v.z = (k0 + col + 2 < HW) ? mb[(size_t)row * HW + k0 + col + 2] : 0.0f;
            v.w = (k0 + col + 3 < HW) ? mb[(size_t)row * HW + k0 + col + 3] : 0.0f;
            *(float4*)&sMask[0][row * MST + col] = v;
        }
        if (w < 3) {
            float4 v;
            v.x = (k0 + col + 0 < HW) ? ib[(size_t)w * HW + k0 + col + 0] : 0.0f;
            v.y = (k0 + col + 1 < HW) ? ib[(size_t)w * HW + k0 + col + 1] : 0.0f;
            v.z = (k0 + col + 2 < HW) ? ib[(size_t)w * HW + k0 + col + 2] : 0.0f;
            v.w = (k0 + col + 3 < HW) ? ib[(size_t)w * HW + k0 + col + 3] : 0.0f;
            *(float4*)&sB[0][w * MST + col] = v;
        }
        __syncthreads();
        compute_tile(sMask[0], sB[0], w, n, half, acc0, acc1);
    }

    // --- cross-wave reduction of D fragments ---------------------------------
    __syncthreads();
    // D layout: VGPR r, lanes 0-15 -> (M=r, N=lane), lanes 16-31 -> (M=8+r, N=lane-16)
    if ((lane & 15) < 4) {
        const int c     = lane & 15;
        const int mhalf = lane >> 4;
        #pragma unroll
        for (int r = 0; r < 8; ++r) {
            const int m0 = r + 8 * mhalf;
            sRed[(w * NMASK + m0) * 4 + c]      = acc0[r];
            sRed[(w * NMASK + m0 + 16) * 4 + c] = acc1[r];
        }
    }
    __syncthreads();

    {
        const int m = tid >> 2;
        const int c = tid & 3;
        const float s = sRed[(0 * NMASK + m) * 4 + c] + sRed[(1 * NMASK + m) * 4 + c]
                      + sRed[(2 * NMASK + m) * 4 + c] + sRed[(3 * NMASK + m) * 4 + c];
        partials[(((size_t)b * nchunks + chunk) * NMASK + m) * 4 + c] = s;
    }
}

// ---------------------------------------------------------------------------
// Epilogue: deterministic partial reduction + normalize + contrast loss
// ---------------------------------------------------------------------------
__global__ void __launch_bounds__(256)
contrast_epilogue(const float* __restrict__ partials,  // [B][nchunks][32][4]
                  const float* __restrict__ valid,     // [B][32]
                  float* __restrict__ out,
                  int B, int nchunks)
{
    __shared__ float sCol[8 * NMASK * 4];
    __shared__ float sNrm[8 * NMASK * 3];
    __shared__ float sVal[8 * NMASK];
    __shared__ float rA[256];
    __shared__ float rB[256];

    const int tid = threadIdx.x;
    const int tot = B * NMASK * 4;

    for (int idx = tid; idx < tot; idx += 256) {
        const int bb  = idx >> 7;
        const int rem = idx & 127;
        float s = 0.0f;
        const float* p = partials + ((size_t)bb * nchunks) * NMASK * 4 + rem;
        for (int ch = 0; ch < nchunks; ++ch)
            s += p[(size_t)ch * NMASK * 4];
        sCol[idx] = s;
    }
    __syncthreads();

    for (int idx = tid; idx < B * NMASK; idx += 256) {
        const float area = fmaxf(sCol[idx * 4 + 3], 1.0f);
        const float cx = sCol[idx * 4 + 0] / area;
        const float cy = sCol[idx * 4 + 1] / area;
        const float cz = sCol[idx * 4 + 2] / area;
        const float nrm = fmaxf(sqrtf(cx * cx + cy * cy + cz * cz), 1e-12f);
        sNrm[idx * 3 + 0] = cx / nrm;
        sNrm[idx * 3 + 1] = cy / nrm;
        sNrm[idx * 3 + 2] = cz / nrm;
        sVal[idx] = valid[idx];
    }
    __syncthreads();

    float contrast = 0.0f, pairs = 0.0f;
    const int npair = B * NMASK * NMASK;
    for (int p = tid; p < npair; p += 256) {
        const int bb = p >> 10;
        const int r  = p & 1023;
        const int i  = r >> 5;
        const int j  = r & 31;
        if (i != j) {
            const float vw = sVal[bb * NMASK + i] * sVal[bb * NMASK + j];
            const float* a = &sNrm[(bb * NMASK + i) * 3];
            const float* c = &sNrm[(bb * NMASK + j) * 3];
            const float dot = a[0] * c[0] + a[1] * c[1] + a[2] * c[2];
            const float sim = dot * TEMP_INV;
            contrast += fmaxf(sim - 0.5f, 0.0f) * vw;
            pairs += vw;
        }
    }
    rA[tid] = contrast;
    rB[tid] = pairs;
    __syncthreads();
    for (int s = 128; s > 0; s >>= 1) {
        if (tid < s) { rA[tid] += rA[tid + s]; rB[tid] += rB[tid + s]; }
        __syncthreads();
    }
    if (tid == 0)
        out[0] = (rA[0] / (rB[0] + 1e-6f)) * 1.0f;  // WEIGHT = 1.0
}

// ---------------------------------------------------------------------------
extern "C" void kernel_launch(void* const* d_in, const int* in_sizes, int n_in,
                              void* d_out, int out_size, void* d_ws, size_t ws_size,
                              hipStream_t stream)
{
    const float* masks  = (const float*)d_in[0];  // pred_masks [B,32,H,W]
    // d_in[1] = target_masks (unused by reference)
    const float* images = (const float*)d_in[2];  // [B,3,H,W]
    const float* valid  = (const float*)d_in[3];  // [B,32]

    const int BN = in_sizes[3];
    const int N  = NMASK;
    const int B  = BN / N;
    const int HW = (int)((size_t)in_sizes[0] / (size_t)BN);

    const int nTiles = (HW + TK - 1) / TK;

    const size_t perChunk = (size_t)B * N * 4 * sizeof(float);
    int nchunks = 96;
    if (perChunk > 0) {
        const size_t maxc = ws_size / perChunk;
        if ((size_t)nchunks > maxc) nchunks = (int)maxc;
    }
    if (nchunks < 1) nchunks = 1;
    if (nchunks > nTiles) nchunks = nTiles;
    const int tilesPerChunk = (nTiles + nchunks - 1) / nchunks;

    float* partials = (float*)d_ws;

    dim3 grid(nchunks, B);
    color_reduce_wmma<<<grid, 128, 0, stream>>>(masks, images, partials,
                                                HW, nTiles, tilesPerChunk, nchunks);
    contrast_epilogue<<<1, 256, 0, stream>>>(partials, valid, (float*)d_out,
                                             B, nchunks);
}